// EntNet_47416438948257
// MI455X (gfx1250) — compile-verified
//
#include <hip/hip_runtime.h>
#include <hip/hip_bf16.h>

// EntNet for MI455X (gfx1250, wave32, WMMA).
// Dims: VOC=32000, MEM=128, NSLOTS=20, NSENT=128 (only first 128 tokens used),
// MAXLEN=32, QLEN=16, ANSLEN=8, B=64.
//
// B-matrices are stored k-pair interleaved in LDS:  P[k2][n][p] = W[n][2*k2+p]
// so each WMMA f32 16x16x4 B-fragment is one aligned ds_load_b64 (no movs).

typedef float v2f __attribute__((ext_vector_type(2)));
typedef float v8f __attribute__((ext_vector_type(8)));

#define DEVINL __device__ __forceinline__

// D = A(16x4 f32) x B(4x16 f32) + C(16x16 f32)  -- precision-exact tensor path.
DEVINL v8f wmma_f32(v2f a, v2f b, v8f c) {
    return __builtin_amdgcn_wmma_f32_16x16x4_f32(
        /*neg_a=*/false, a, /*neg_b=*/false, b,
        /*c_mod=*/(short)0, c, /*reuse_a=*/false, /*reuse_b=*/false);
}

// -------------------------------------------------------------------------
// K1: per-batch prep.  Grid=64 (b), Block=256 (8 waves).
// - gather s[b,t,:] = E[ids[b*4096+t]] for t<128, stage in LDS + spill to ws
// - Ws = s @ Ww^T + Wb   -> ws            (WMMA f32 16x16x4)
// - SK = s @ slot_keys^T -> ws (j padded to 32)
// -------------------------------------------------------------------------
__global__ void k1_prep(const int* __restrict__ ids, const float* __restrict__ E,
                        const float* __restrict__ Ww, const float* __restrict__ Wb,
                        const float* __restrict__ skeys,
                        float* __restrict__ s_ws, float* __restrict__ Ws_ws,
                        float* __restrict__ sk_ws) {
    extern __shared__ float sm[];
    float* WwP = sm;              // 64*128*2 : WwP[k2*256 + n*2 + p] = Ww[n*128 + 2*k2+p]
    float* skP = WwP + 16384;     // 64*32*2  : skP[k2*64 + j*2 + p] = skeys[j*128 + 2*k2+p]
    float* sS  = skP + 4096;      // 128*128  : s rows for this batch
    float* wbL = sS + 16384;      // 128

    const int tid = threadIdx.x;
    const int b   = blockIdx.x;

    for (int i = tid; i < 16384; i += 256) {
        int n = i >> 7, m = i & 127;
        WwP[(m >> 1) * 256 + n * 2 + (m & 1)] = Ww[i];
    }
    for (int i = tid; i < 4096; i += 256) {
        int j = i >> 7, m = i & 127;
        float v = (j < 20) ? skeys[j * 128 + m] : 0.0f;
        skP[(m >> 1) * 64 + j * 2 + (m & 1)] = v;
    }
    if (tid < 128) wbL[tid] = Wb[tid];

    const int* tokp = ids + (size_t)b * 4096;   // only first 128 tokens used
    for (int i = tid; i < 128 * 32; i += 256) {
        int t = i >> 5, c4 = (i & 31) * 4;
        int tok = tokp[t];
        float4 v = *(const float4*)(E + (size_t)tok * 128 + c4);
        *(float4*)(sS + t * 128 + c4) = v;
        *(float4*)(s_ws + (size_t)b * 16384 + t * 128 + c4) = v;
    }
    __syncthreads();

    const int w = tid >> 5, l = tid & 31;
    const int lm = l & 15, lh = l >> 4;
    const int Mbase = w * 16;                   // wave w owns rows t = Mbase..+15

    // ---- Ws GEMM: 8 N-tiles, K=128 (32 k-steps of 4) ----
    v8f acc[8] = {};
    for (int kt = 0; kt < 32; ++kt) {
        const int kb = kt * 4;
        const int k2 = kt * 2 + lh;             // row-pair index (kb + 2*lh)/2
        v2f a = *(const v2f*)(sS + (Mbase + lm) * 128 + kb + 2 * lh);
#pragma unroll
        for (int nt = 0; nt < 8; ++nt) {
            v2f bb = *(const v2f*)(WwP + k2 * 256 + (nt * 16 + lm) * 2);
            acc[nt] = wmma_f32(a, bb, acc[nt]);
        }
    }
#pragma unroll
    for (int nt = 0; nt < 8; ++nt) {
        const int Nb = nt * 16;
        const float wb = wbL[Nb + lm];
#pragma unroll
        for (int r = 0; r < 8; ++r) {
            const int trow = Mbase + r + 8 * lh;
            Ws_ws[(size_t)b * 16384 + trow * 128 + Nb + lm] = acc[nt][r] + wb;
        }
    }

    // ---- SK GEMM: 2 N-tiles (j padded to 32) ----
    v8f ak[2] = {};
    for (int kt = 0; kt < 32; ++kt) {
        const int kb = kt * 4;
        const int k2 = kt * 2 + lh;
        v2f a = *(const v2f*)(sS + (Mbase + lm) * 128 + kb + 2 * lh);
#pragma unroll
        for (int nt = 0; nt < 2; ++nt) {
            v2f bb = *(const v2f*)(skP + k2 * 64 + (nt * 16 + lm) * 2);
            ak[nt] = wmma_f32(a, bb, ak[nt]);
        }
    }
#pragma unroll
    for (int nt = 0; nt < 2; ++nt)
#pragma unroll
        for (int r = 0; r < 8; ++r) {
            const int trow = Mbase + r + 8 * lh;
            sk_ws[(size_t)b * 4096 + trow * 32 + nt * 16 + lm] = ak[nt][r];
        }
}

// -------------------------------------------------------------------------
// K2: the scan.  Grid=20 (one WG per slot j), Block=256 (8 waves).
// mem_j (64x128) lives in LDS for all 128 steps; Uw (pair-interleaved) in LDS.
// Per step: gate dot, WMMA GEMM h_cand, gated update, per-slot L2 norm.
// -------------------------------------------------------------------------
__global__ void k2_scan(const float* __restrict__ Uw, const float* __restrict__ Ub,
                        const float* __restrict__ Vw, const float* __restrict__ Vb,
                        const float* __restrict__ skeys,
                        const float* __restrict__ s_ws, const float* __restrict__ Ws_ws,
                        const float* __restrict__ sk_ws, float* __restrict__ h_ws) {
    extern __shared__ float sm[];
    float* UwP  = sm;              // 16384 : UwP[k2*256 + n*2 + p] = Uw[n*128 + 2*k2+p]
    float* mem  = UwP + 16384;     // 8192  : mem[b][m]
    float* sT   = mem + 8192;      // 8192  : s_t rows
    float* wsT  = sT + 8192;       // 8192  : Ws_t rows
    float* ubvk = wsT + 8192;      // 128   : Ub + Vkey_j
    float* keyL = ubvk + 128;      // 128
    float* gL   = keyL + 128;      // 64
    float* gp   = gL + 64;         // 256
    float* wred = gp + 256;        // 8
    float* nrmL = wred + 8;        // 1

    const int tid = threadIdx.x;
    const int j   = blockIdx.x;

    for (int i = tid; i < 16384; i += 256) {
        int n = i >> 7, m = i & 127;
        UwP[(m >> 1) * 256 + n * 2 + (m & 1)] = Uw[i];
    }
    if (tid < 128) keyL[tid] = skeys[j * 128 + tid];
    for (int i = tid; i < 8192; i += 256) mem[i] = 0.0f;
    __syncthreads();
    if (tid < 128) {                 // Vkey_j[n] = Vw[n,:]·key_j + Vb[n]; fold Ub
        float a = Vb[tid];
        const float* vr = Vw + tid * 128;
        for (int m = 0; m < 128; ++m) a += vr[m] * keyL[m];
        ubvk[tid] = a + Ub[tid];
    }
    __syncthreads();

    const int w = tid >> 5, l = tid & 31;
    const int lm = l & 15, lh = l >> 4;
    const int Mbase  = (w >> 1) * 16;     // 2 waves per 16-row M-tile
    const int ntbase = (w & 1) * 4;       // 4 N-tiles each

    for (int t = 0; t < 128; ++t) {
        // stage s_t and Ws_t (coalesced, L2-resident)
        {
            const int r = tid >> 2, part = (tid & 3) * 32;
            const float* sp = s_ws  + (size_t)r * 16384 + t * 128 + part;
            const float* wp = Ws_ws + (size_t)r * 16384 + t * 128 + part;
            float* sd = sT  + r * 128 + part;
            float* wd = wsT + r * 128 + part;
#pragma unroll
            for (int k = 0; k < 8; ++k) {
                *(float4*)(sd + 4 * k) = *(const float4*)(sp + 4 * k);
                *(float4*)(wd + 4 * k) = *(const float4*)(wp + 4 * k);
            }
        }
        __syncthreads();

        // gate partial dots: 4 threads per batch row
        {
            const int bb = tid >> 2, part = (tid & 3) * 32;
            const float* sv = sT  + bb * 128 + part;
            const float* mv = mem + bb * 128 + part;
            float a = 0.0f;
#pragma unroll
            for (int k = 0; k < 32; ++k) a += sv[k] * mv[k];
            gp[tid] = a;
        }
        __syncthreads();
        if (tid < 64) {
            float d = gp[4 * tid] + gp[4 * tid + 1] + gp[4 * tid + 2] + gp[4 * tid + 3]
                    + sk_ws[(size_t)tid * 4096 + t * 32 + j];
            gL[tid] = 1.0f / (1.0f + __expf(-d));
        }

        // h_cand GEMM: mem(64x128) @ Uw^T -> register accumulators (reads mem only)
        v8f acc[4] = {};
        for (int kt = 0; kt < 32; ++kt) {
            const int kb = kt * 4;
            const int k2 = kt * 2 + lh;
            v2f a = *(const v2f*)(mem + (Mbase + lm) * 128 + kb + 2 * lh);
#pragma unroll
            for (int i = 0; i < 4; ++i) {
                v2f bb = *(const v2f*)(UwP + k2 * 256 + ((ntbase + i) * 16 + lm) * 2);
                acc[i] = wmma_f32(a, bb, acc[i]);
            }
        }
        __syncthreads();   // all reads of mem complete; gL visible

        // gated update in place + local sum of squares
        float sq = 0.0f;
#pragma unroll
        for (int i = 0; i < 4; ++i) {
            const int n = (ntbase + i) * 16 + lm;
            const float uv = ubvk[n];
#pragma unroll
            for (int r = 0; r < 8; ++r) {
                const int bb = Mbase + r + 8 * lh;
                float h = acc[i][r] + uv + wsT[bb * 128 + n];
                h = fmaxf(h, 0.0f);
                float nv = mem[bb * 128 + n] + gL[bb] * h;
                mem[bb * 128 + n] = nv;
                sq += nv * nv;
            }
        }
#pragma unroll
        for (int off = 16; off > 0; off >>= 1) sq += __shfl_xor(sq, off, 32);
        if (l == 0) wred[w] = sq;
        __syncthreads();
        if (tid == 0) {
            float tot = 0.0f;
#pragma unroll
            for (int i = 0; i < 8; ++i) tot += wred[i];
            nrmL[0] = 1.0f / sqrtf(tot);
        }
        __syncthreads();
        const float inv = nrmL[0];
        for (int i = tid; i < 8192; i += 256) mem[i] *= inv;
        __syncthreads();
    }

    for (int i = tid; i < 8192; i += 256) {
        const int bb = i >> 7, m = i & 127;
        h_ws[((size_t)bb * 20 + j) * 128 + m] = mem[i];
    }
}

// -------------------------------------------------------------------------
// K3: readout.  Grid=64 (b), Block=128.
// -------------------------------------------------------------------------
__global__ void k3_readout(const int* __restrict__ question, const int* __restrict__ ans,
                           const float* __restrict__ E, const float* __restrict__ Hw,
                           const float* __restrict__ Hb, const float* __restrict__ h_ws,
                           float* __restrict__ out) {
    __shared__ float hL[20][128];
    __shared__ float qL[128], a1L[128], a2L[128], uL[128], rL[128];
    __shared__ float pL[20];
    const int b = blockIdx.x, tid = threadIdx.x;

    for (int jj = 0; jj < 20; ++jj)
        hL[jj][tid] = h_ws[((size_t)b * 20 + jj) * 128 + tid];
    {
        float q = 0.f, a1 = 0.f, a2 = 0.f;
        for (int i = 0; i < 16; ++i) q += E[(size_t)question[b * 16 + i] * 128 + tid];
        for (int i = 0; i < 8; ++i) {
            a1 += E[(size_t)ans[(b * 8 + i) * 2 + 0] * 128 + tid];
            a2 += E[(size_t)ans[(b * 8 + i) * 2 + 1] * 128 + tid];
        }
        qL[tid] = q * (1.0f / 16.0f);
        a1L[tid] = a1 * 0.125f;
        a2L[tid] = a2 * 0.125f;
    }
    __syncthreads();
    if (tid < 20) {
        float d = 0.f;
        for (int m = 0; m < 128; ++m) d += hL[tid][m] * qL[m];
        pL[tid] = d;
    }
    __syncthreads();
    if (tid == 0) {
        float mx = -1e30f;
        for (int jj = 0; jj < 20; ++jj) mx = fmaxf(mx, pL[jj]);
        float s = 0.f;
        for (int jj = 0; jj < 20; ++jj) { float e = __expf(pL[jj] - mx); pL[jj] = e; s += e; }
        float is = 1.0f / s;
        for (int jj = 0; jj < 20; ++jj) pL[jj] *= is;
    }
    __syncthreads();
    {
        float u = 0.f;
        for (int jj = 0; jj < 20; ++jj) u += pL[jj] * hL[jj][tid];
        uL[tid] = u;
    }
    __syncthreads();
    {
        float r = Hb[tid] + qL[tid];
        const float* hr = Hw + tid * 128;
        for (int m = 0; m < 128; ++m) r += hr[m] * uL[m];
        rL[tid] = fmaxf(r, 0.0f);
    }
    __syncthreads();
    if (tid == 0) {
        float y1 = 0.f, y2 = 0.f;
        for (int m = 0; m < 128; ++m) { y1 += rL[m] * a1L[m]; y2 += rL[m] * a2L[m]; }
        float mx = fmaxf(y1, y2);
        float e1 = __expf(y1 - mx), e2 = __expf(y2 - mx);
        float is = 1.0f / (e1 + e2);
        out[b * 2 + 0] = e1 * is;
        out[b * 2 + 1] = e2 * is;
    }
}

extern "C" void kernel_launch(void* const* d_in, const int* in_sizes, int n_in,
                              void* d_out, int out_size, void* d_ws, size_t ws_size,
                              hipStream_t stream) {
    const int*   ids      = (const int*)d_in[0];
    const int*   question = (const int*)d_in[1];
    const int*   ans      = (const int*)d_in[2];
    const float* E        = (const float*)d_in[3];
    const float* Uw       = (const float*)d_in[4];
    const float* Ub       = (const float*)d_in[5];
    const float* Vw       = (const float*)d_in[6];
    const float* Vb       = (const float*)d_in[7];
    const float* Ww       = (const float*)d_in[8];
    const float* Wb       = (const float*)d_in[9];
    const float* skeys    = (const float*)d_in[10];
    const float* Hw       = (const float*)d_in[11];
    const float* Hb       = (const float*)d_in[12];
    float* out = (float*)d_out;

    float* ws    = (float*)d_ws;
    float* s_ws  = ws;                       // 64*128*128 = 1048576
    float* Ws_ws = ws + 1048576;             // 1048576
    float* sk_ws = ws + 2097152;             // 64*128*32 = 262144
    float* h_ws  = ws + 2359296;             // 64*20*128 = 163840   (total ~10.1 MB)

    const size_t lds1 = (size_t)(16384 + 4096 + 16384 + 128) * sizeof(float);
    k1_prep<<<64, 256, lds1, stream>>>(ids, E, Ww, Wb, skeys, s_ws, Ws_ws, sk_ws);

    const size_t lds2 = (size_t)(16384 + 8192 * 3 + 128 + 128 + 64 + 256 + 8 + 8) * sizeof(float);
    k2_scan<<<20, 256, lds2, stream>>>(Uw, Ub, Vw, Vb, skeys, s_ws, Ws_ws, sk_ws, h_ws);

    k3_readout<<<64, 128, 0, stream>>>(question, ans, E, Hw, Hb, h_ws, out);
}